// VectorQuantizer_13383118094409
// MI455X (gfx1250) — compile-verified
//
#include <hip/hip_runtime.h>
#include <hip/hip_bf16.h>

typedef __attribute__((ext_vector_type(2))) float v2f;
typedef __attribute__((ext_vector_type(8))) float v8f;
typedef int b128_t __attribute__((vector_size(16)));   // builtin's payload type

#define NUM_CODES   8192
#define EMBED_DIM   256
#define BATCH       16
#define HW          1024
#define NPOS        (BATCH * HW)                 // 16384 positions
#define ZQ_ELEMS    (BATCH * EMBED_DIM * HW)     // 4194304
#define BETA        0.25f
#define LDS_STRIDE  264                          // 256 + 8 pad -> 16B-aligned rows for B128 async
#define KSTEPS      (EMBED_DIM / 4)              // 64 WMMA K-steps (f32 WMMA K=4)
#define NTILES      (NUM_CODES / 16)             // 512 code tiles
#define NSLICES     4
#define NT_SLICE    (NTILES / NSLICES)           // 128 tiles per N-slice

#define AS1 __attribute__((address_space(1)))
#define AS3 __attribute__((address_space(3)))

#if __has_builtin(__builtin_amdgcn_global_load_async_to_lds_b128)
#define HAVE_ASYNC_LDS 1
#else
#define HAVE_ASYNC_LDS 0
#endif

#if __has_builtin(__builtin_amdgcn_s_wait_asynccnt)
#define WAIT_ASYNC(n) __builtin_amdgcn_s_wait_asynccnt(n)
#else
#define WAIT_ASYNC(n) asm volatile("s_wait_asynccnt %0" ::"n"(n) : "memory")
#endif

// ---------------------------------------------------------------------------
// Kernel 0: zero scratch accumulators (harness does not re-zero between runs)
// ---------------------------------------------------------------------------
__global__ void vq_zero(unsigned* __restrict__ presence, float* __restrict__ loss_acc) {
  int t = blockIdx.x * blockDim.x + threadIdx.x;
  if (t < BATCH * NUM_CODES / 32) presence[t] = 0u;
  if (t == 0) *loss_acc = 0.f;
}

// ---------------------------------------------------------------------------
// Kernel 1: per-code squared norms ||e||^2 (one wave per code, coalesced)
// ---------------------------------------------------------------------------
__global__ void vq_norms(const float* __restrict__ cb, float* __restrict__ norms) {
  const int lane = threadIdx.x & 31;
  const int wid  = threadIdx.x >> 5;
  const int code = blockIdx.x * 8 + wid;
  const float* row = cb + (size_t)code * EMBED_DIM;
  float s = 0.f;
#pragma unroll
  for (int i = 0; i < EMBED_DIM / 32; ++i) {
    float v = row[lane + i * 32];
    s = fmaf(v, v, s);
  }
#pragma unroll
  for (int off = 16; off >= 1; off >>= 1) s += __shfl_xor(s, off, 32);
  if (lane == 0) norms[code] = s;
}

// ---------------------------------------------------------------------------
// Cooperative staging of one 16-code x 256-f32 tile into (padded) LDS.
// Async path: GLOBAL_LOAD_ASYNC_TO_LDS_B128 (ASYNCcnt, no VGPR round-trip).
// Address-space casts go through integers: flat==global addr for AS1 (ISA
// aperture rule), and addr[31:0] of a flat LDS pointer is the LDS offset.
// ---------------------------------------------------------------------------
__device__ __forceinline__ void stage_tile(const float* __restrict__ src, float* dst) {
#pragma unroll
  for (int i = 0; i < 4; ++i) {
    const int f4 = (int)threadIdx.x + i * 256;   // 1024 float4 per tile
    const int r  = f4 >> 6;                      // 64 float4 per code row
    const int c  = (f4 & 63) << 2;
#if HAVE_ASYNC_LDS
    AS1 b128_t* g = (AS1 b128_t*)(unsigned long long)(src + (size_t)f4 * 4);
    AS3 b128_t* l = (AS3 b128_t*)(unsigned)(unsigned long long)(dst + r * LDS_STRIDE + c);
    __builtin_amdgcn_global_load_async_to_lds_b128(g, l, 0, 0);
#else
    float4 v = ((const float4*)src)[f4];
    *(float4*)(dst + r * LDS_STRIDE + c) = v;    // 16B aligned (stride 264)
#endif
  }
}

// ---------------------------------------------------------------------------
// Kernel 2: WMMA f32 GEMM (scores = z . e) fused with running argmin of
// d = ||e||^2 - 2*score.  Wave = one 16-position M-tile; block of 8 waves
// shares double-buffered LDS code tiles; grid.y slices the 8192 codes by 4.
// A slab register-resident (64 x v2f); two interleaved accumulator chains.
// ---------------------------------------------------------------------------
__global__ void __launch_bounds__(256, 1)
vq_argmin(const float* __restrict__ z, const float* __restrict__ cb,
          const float* __restrict__ norms, float* __restrict__ pminv,
          int* __restrict__ pminidx) {
  __shared__ float bt[2][16 * LDS_STRIDE];       // 2 x 16.5 KB

  const int lane  = threadIdx.x & 31;
  const int wid   = threadIdx.x >> 5;
  const int tile  = blockIdx.x * 8 + wid;        // 0..1023 M-tiles
  const int slice = blockIdx.y;                  // 0..3 N-slices
  const int mbase = tile * 16;
  const int b     = mbase >> 10;                 // 16 rows never straddle a batch
  const int hw    = mbase & 1023;

  // --- preload A fragments in V_WMMA_F32_16X16X4_F32 layout ---------------
  // lane L: M = L&15 ; VGPR0 holds K = 2*(L>>4), VGPR1 holds K+1
  const int am = lane & 15;
  const int kb = (lane >> 4) << 1;
  const float* abase = z + (size_t)b * EMBED_DIM * HW + hw + am;
  v2f afrag[KSTEPS];
#pragma unroll
  for (int ks = 0; ks < KSTEPS; ++ks) {
    afrag[ks].x = abase[(size_t)(ks * 4 + kb)     * HW];
    afrag[ks].y = abase[(size_t)(ks * 4 + kb + 1) * HW];
  }

  float minv[8];
  int   minidx[8];
#pragma unroll
  for (int j = 0; j < 8; ++j) { minv[j] = 3.4e38f; minidx[j] = 0; }

  const int ncol = lane & 15;          // C/D layout: lane -> column N = lane&15
  const int bko  = (lane >> 4) << 1;   // B frag K sub-offset (mirror of A)
  const int nt0  = slice * NT_SLICE;

  stage_tile(cb + (size_t)nt0 * 16 * EMBED_DIM, bt[0]);   // prologue: tile 0

  for (int t = 0; t < NT_SLICE; ++t) {
    const int nt = nt0 + t;
    __syncthreads();                   // all waves done with buffer (t+1)&1
    if (t + 1 < NT_SLICE)
      stage_tile(cb + (size_t)(nt + 1) * 16 * EMBED_DIM, bt[(t + 1) & 1]);
#if HAVE_ASYNC_LDS
    if (t + 1 < NT_SLICE) { WAIT_ASYNC(4); } else { WAIT_ASYNC(0); }
#endif
    __syncthreads();                   // tile t fully resident in LDS
    const float* buf = bt[t & 1];

    const float nrm = norms[nt * 16 + ncol];

    v8f c0 = {0.f, 0.f, 0.f, 0.f, 0.f, 0.f, 0.f, 0.f};
    v8f c1 = {0.f, 0.f, 0.f, 0.f, 0.f, 0.f, 0.f, 0.f};
#pragma unroll
    for (int ks = 0; ks < KSTEPS / 2; ++ks) {    // two independent chains
      v2f b0 = *(const v2f*)&buf[ncol * LDS_STRIDE + ks * 4 + bko];
      v2f b1 = *(const v2f*)&buf[ncol * LDS_STRIDE + (ks + 32) * 4 + bko];
      c0 = __builtin_amdgcn_wmma_f32_16x16x4_f32(false, afrag[ks], false, b0,
                                                 (short)0, c0, false, false);
      c1 = __builtin_amdgcn_wmma_f32_16x16x4_f32(false, afrag[ks + 32], false, b1,
                                                 (short)0, c1, false, false);
    }

    const int n = nt * 16 + ncol;
#pragma unroll
    for (int j = 0; j < 8; ++j) {
      float d = fmaf(-2.f, c0[j] + c1[j], nrm);
      if (d < minv[j]) { minv[j] = d; minidx[j] = n; }  // ascending nt keeps first-min
    }
  }

  // cross-lane argmin over 16 column slots (xor<16 stays inside the group)
#pragma unroll
  for (int j = 0; j < 8; ++j) {
    float v = minv[j];
    int   i = minidx[j];
#pragma unroll
    for (int off = 1; off < 16; off <<= 1) {
      float ov = __shfl_xor(v, off, 32);
      int   oi = __shfl_xor(i, off, 32);
      if (ov < v || (ov == v && oi < i)) { v = ov; i = oi; }
    }
    minv[j] = v; minidx[j] = i;
  }

  if ((lane & 15) == 0) {
    const int rb = (lane >> 4) << 3;             // rows 0-7 or 8-15
#pragma unroll
    for (int j = 0; j < 8; ++j) {
      pminv  [slice * NPOS + mbase + rb + j] = minv[j];
      pminidx[slice * NPOS + mbase + rb + j] = minidx[j];
    }
  }
}

// ---------------------------------------------------------------------------
// Kernel 2b: merge the 4 N-slice partial argmins (slice order = ascending idx,
// so strict < preserves first-occurrence semantics of jnp.argmin)
// ---------------------------------------------------------------------------
__global__ void vq_merge(const float* __restrict__ pminv, const int* __restrict__ pminidx,
                         int* __restrict__ indices) {
  const int m = blockIdx.x * 256 + threadIdx.x;
  float v = pminv[m];
  int   i = pminidx[m];
#pragma unroll
  for (int s = 1; s < NSLICES; ++s) {
    float ov = pminv[s * NPOS + m];
    int   oi = pminidx[s * NPOS + m];
    if (ov < v) { v = ov; i = oi; }
  }
  indices[m] = i;
}

// ---------------------------------------------------------------------------
// Kernel 3: gather z_q = codebook[idx], write [B,C,H,W] output + index(f32),
// accumulate loss sum, mark presence bitmap.
// ---------------------------------------------------------------------------
__global__ void vq_gather(const float* __restrict__ z, const float* __restrict__ cb,
                          const int* __restrict__ indices, float* __restrict__ out,
                          unsigned* __restrict__ presence, float* __restrict__ loss_acc) {
  __shared__ float wsum[8];
  const int lane = threadIdx.x & 31;
  const int ty   = threadIdx.x >> 5;
  const int pos0 = blockIdx.x * 32;
  const int b    = pos0 >> 10;
  const int hw   = (pos0 & 1023) + lane;
  const int m    = pos0 + lane;
  const int idx  = indices[m];

  if (ty == 0) {
    out[ZQ_ELEMS + m] = (float)idx;              // index output (as f32)
    atomicOr(&presence[b * (NUM_CODES / 32) + (idx >> 5)], 1u << (idx & 31));
  }

  const float* crow = cb + (size_t)idx * EMBED_DIM;
  float lsum = 0.f;
#pragma unroll
  for (int i = 0; i < EMBED_DIM / 8; ++i) {
    const int ch = ty * 32 + i;
    float zq = crow[ch];                          // L2-resident gather
    size_t off = (size_t)b * EMBED_DIM * HW + (size_t)ch * HW + hw;
    float zl = z[off];
    out[off] = zq;                                // straight-through fwd == z_q
    float dd = zq - zl;
    lsum = fmaf(dd, dd, lsum);
  }
#pragma unroll
  for (int off = 16; off >= 1; off >>= 1) lsum += __shfl_xor(lsum, off, 32);
  if (lane == 0) wsum[ty] = lsum;
  __syncthreads();
  if (threadIdx.x == 0) {
    float tt = 0.f;
#pragma unroll
    for (int i = 0; i < 8; ++i) tt += wsum[i];
    atomicAdd(loss_acc, tt);
  }
}

// ---------------------------------------------------------------------------
// Kernel 4: finalize loss + diversity scalars
// ---------------------------------------------------------------------------
__global__ void vq_finalize(const unsigned* __restrict__ presence,
                            const float* __restrict__ loss_acc, float* __restrict__ out) {
  __shared__ int wsum[8];
  const int lane = threadIdx.x & 31;
  const int ty   = threadIdx.x >> 5;
  int cnt = 0;
  for (int i = threadIdx.x; i < BATCH * NUM_CODES / 32; i += 256)
    cnt += __popc(presence[i]);
#pragma unroll
  for (int off = 16; off >= 1; off >>= 1) cnt += __shfl_xor(cnt, off, 32);
  if (lane == 0) wsum[ty] = cnt;
  __syncthreads();
  if (threadIdx.x == 0) {
    int t = 0;
#pragma unroll
    for (int i = 0; i < 8; ++i) t += wsum[i];
    out[ZQ_ELEMS + NPOS]     = (*loss_acc) * (1.f + BETA) / (float)ZQ_ELEMS;
    out[ZQ_ELEMS + NPOS + 1] = (float)t / (float)NPOS;
  }
}

// ---------------------------------------------------------------------------
extern "C" void kernel_launch(void* const* d_in, const int* in_sizes, int n_in,
                              void* d_out, int out_size, void* d_ws, size_t ws_size,
                              hipStream_t stream) {
  const float* z  = (const float*)d_in[0];   // [16, 256, 32, 32]
  const float* cb = (const float*)d_in[1];   // [8192, 256]
  float* out = (float*)d_out;                // zq(4194304) | index(16384) | loss | diversity
  char* ws = (char*)d_ws;

  size_t o = 0;
  float*    norms    = (float*)(ws + o);  o += (size_t)NUM_CODES * 4;            // 32 KB
  unsigned* presence = (unsigned*)(ws + o); o += (size_t)BATCH * NUM_CODES / 8;  // 16 KB
  float*    loss_acc = (float*)(ws + o);  o += 16;
  int*      indices  = (int*)(ws + o);    o += (size_t)NPOS * 4;                 // 64 KB
  float*    pminv    = (float*)(ws + o);  o += (size_t)NSLICES * NPOS * 4;       // 256 KB
  int*      pminidx  = (int*)(ws + o);    o += (size_t)NSLICES * NPOS * 4;       // 256 KB

  vq_zero    <<<16,                    256, 0, stream>>>(presence, loss_acc);
  vq_norms   <<<NUM_CODES / 8,         256, 0, stream>>>(cb, norms);
  vq_argmin  <<<dim3(NPOS / 128, NSLICES), 256, 0, stream>>>(z, cb, norms, pminv, pminidx);
  vq_merge   <<<NPOS / 256,            256, 0, stream>>>(pminv, pminidx, indices);
  vq_gather  <<<NPOS / 32,             256, 0, stream>>>(z, cb, indices, out, presence, loss_acc);
  vq_finalize<<<1,                     256, 0, stream>>>(presence, loss_acc, out);
}